// KernelShiftedPrediction_5188320493686
// MI455X (gfx1250) — compile-verified
//
#include <hip/hip_runtime.h>
#include <stdint.h>

// Problem constants (from reference): predicted (S,B,H,W), target (B,S,H,W), KSZ=1
#define S_DIM 10
#define B_DIM 8
#define H_DIM 512
#define W_DIM 512
#define TILE_W 128
#define TILE_H 8
#define LDS_W (TILE_W + 2)   // 130 floats per staged row (1-px halo each side)
#define LDS_H (TILE_H + 2)   // 10 staged rows

typedef unsigned int u32x4 __attribute__((ext_vector_type(4)));
typedef int          i32x4 __attribute__((ext_vector_type(4)));
typedef int          i32x8 __attribute__((ext_vector_type(8)));

__global__ __launch_bounds__(256) void shifted_pred_kernel(
    const float* __restrict__ pred,   // (S,B,H,W)
    const float* __restrict__ tgt,    // (B,S,H,W)
    float* __restrict__ out)          // (S,B,H,W)
{
  // Single LDS allocation -> placed at LDS offset 0 (used by the TDM descriptor).
  __shared__ float tile[LDS_H * LDS_W];   // 5200 B

  const int tx_blk = blockIdx.x * TILE_W;
  const int ty_blk = blockIdx.y * TILE_H;
  const int z      = blockIdx.z;               // z = s*B + b
  const int s      = z / B_DIM;
  const int b      = z - s * B_DIM;

  const long long predPlane = (long long)z * (long long)(H_DIM * W_DIM);
  const long long tgtPlane  = (long long)(b * S_DIM + s) * (long long)(H_DIM * W_DIM);

  // Clamp the halo window fully inside the plane; shift-validity is checked
  // arithmetically below, so clamp-induced reuse rows/cols are never selected.
  int gx0 = tx_blk - 1;
  if (gx0 < 0) gx0 = 0;
  if (gx0 > W_DIM - LDS_W) gx0 = W_DIM - LDS_W;
  int gy0 = ty_blk - 1;
  if (gy0 < 0) gy0 = 0;
  if (gy0 > H_DIM - LDS_H) gy0 = H_DIM - LDS_H;

  // ---- Stage LDS_H x LDS_W halo'd tile of `pred` into LDS --------------------
#if __has_builtin(__builtin_amdgcn_tensor_load_to_lds)
  if (threadIdx.x < 32) {   // one wave issues the DMA; TENSORcnt is per-wave
    const unsigned long long gaddr =
        (unsigned long long)(uintptr_t)(pred + predPlane + (long long)gy0 * W_DIM + gx0);

    // D# group 0 (ISA 8.3): count=1 | lds_addr | global_addr[56:0] | type=2
    u32x4 g0;
    g0.x = 1u;                                     // count=1, user descriptor
    g0.y = 0u;                                     // lds_addr = 0 (tile[] base)
    g0.z = (unsigned)(gaddr & 0xFFFFFFFFull);      // global_addr[31:0]
    g0.w = ((unsigned)(gaddr >> 32) & 0x01FFFFFFu) // global_addr[56:32]
           | 0x80000000u;                          // type = 2 ("image")

    // D# group 1 (ISA 8.4): mask=0, data_size=4B, dims/strides for a 2D tile
    i32x8 g1;
    g1[0] = 2 << 16;                               // data_size = 2 (4 bytes)
    g1[1] = (int)((unsigned)W_DIM << 16);          // tensor_dim0[15:0]  @ [31:16]
    g1[2] = (int)((unsigned)H_DIM << 16);          // tensor_dim1[15:0]  @ [31:16]
    g1[3] = (int)((unsigned)LDS_W << 16);          // tile_dim0          @ [31:16]
    g1[4] = LDS_H;                                 // tile_dim1 (tile_dim2 = 0)
    g1[5] = W_DIM;                                 // tensor_dim0_stride[31:0]
    g1[6] = 0;                                     // stride0[47:32]=0, stride1 lo=0
    g1[7] = 0;

    i32x4 gz = {0, 0, 0, 0};                       // groups 2/3 unused (2D tensor)
#if defined(__clang_major__) && __clang_major__ >= 23
    i32x8 gz8 = {0, 0, 0, 0, 0, 0, 0, 0};
    __builtin_amdgcn_tensor_load_to_lds(g0, g1, gz, gz, gz8, 0);
#else
    __builtin_amdgcn_tensor_load_to_lds(g0, g1, gz, gz, 0);
#endif
    __builtin_amdgcn_s_wait_tensorcnt(0);          // data landed in LDS
  }
#else
  // Fallback: cooperative staged load (always in-bounds thanks to clamping).
  for (int i = threadIdx.x; i < LDS_H * LDS_W; i += 256) {
    const int r = i / LDS_W;
    const int c = i - r * LDS_W;
    tile[i] = pred[predPlane + (long long)(gy0 + r) * W_DIM + (gx0 + c)];
  }
#endif

  // The TDM writes LDS through the descriptor (lds_addr is an integer, not a
  // pointer the compiler can see), so alias analysis would otherwise conclude
  // `tile` is never stored and fold every LDS load to undef — which deleted
  // the whole compute phase last round. Escape the pointer with a memory
  // clobber so the staged data is considered live.
  asm volatile("" : : "v"(&tile[0]) : "memory");

  __syncthreads();

  // ---- Compute: 4 pixels per thread, float4 target load / output store ------
  const int lane = threadIdx.x & 31;
  const int row  = threadIdx.x >> 5;        // 0..7
  const int h    = ty_blk + row;
  const int w0   = tx_blk + lane * 4;

  const float4 t4 = *(const float4*)(tgt + tgtPlane + (long long)h * W_DIM + w0);
  const float tt[4] = {t4.x, t4.y, t4.z, t4.w};

  const int lr  = h  - gy0;                 // LDS row of (h, *)
  const int lc0 = w0 - gx0;                 // LDS col of w0

  // Register-cache the 3x6 neighborhood shared by this thread's 4 pixels.
  float nb[3][6];
#pragma unroll
  for (int x = 0; x < 3; ++x)
#pragma unroll
    for (int y = 0; y < 6; ++y)
      nb[x][y] = tile[(lr - 1 + x) * LDS_W + (lc0 - 1 + y)];

  float4 o;
  float* op = &o.x;
#pragma unroll
  for (int j = 0; j < 4; ++j) {
    const int   w = w0 + j;
    const float t = tt[j];

    // Candidate 0: shift (0,0) — always valid (ties resolve to it, like argmin).
    float bestc = nb[1][j + 1];
    float bestl = fabsf(t - bestc);

#pragma unroll
    for (int x = -1; x <= 1; ++x) {
      const bool vr = (unsigned)(h + x) < (unsigned)H_DIM;
#pragma unroll
      for (int y = -1; y <= 1; ++y) {
        const bool  v = vr && ((unsigned)(w + y) < (unsigned)W_DIM);
        const float c = nb[x + 1][j + y + 1];
        const float l = fabsf(t - c);
        const bool take = v && (l < bestl);   // strict '<' => first-min wins
        bestl = take ? l : bestl;
        bestc = take ? c : bestc;
      }
    }
    op[j] = bestc;
  }

  *(float4*)(out + predPlane + (long long)h * W_DIM + w0) = o;
}

extern "C" void kernel_launch(void* const* d_in, const int* in_sizes, int n_in,
                              void* d_out, int out_size, void* d_ws, size_t ws_size,
                              hipStream_t stream) {
  const float* pred = (const float*)d_in[0];   // (S,B,H,W) fp32
  const float* tgt  = (const float*)d_in[1];   // (B,S,H,W) fp32
  // d_in[2] (mask) is unused: reference uses weights = ones_like(target).
  float* out = (float*)d_out;                  // (S,B,H,W) fp32

  dim3 grid(W_DIM / TILE_W, H_DIM / TILE_H, S_DIM * B_DIM);  // (4, 64, 80)
  shifted_pred_kernel<<<grid, 256, 0, stream>>>(pred, tgt, out);
}